// QNetwork_76871324663842
// MI455X (gfx1250) — compile-verified
//
#include <hip/hip_runtime.h>

// ---------------------------------------------------------------------------
// Problem constants (from reference): B=65536, D=128, H=256, O=8, E=8
// q = relu(x @ Ws + bs) @ (W1[p]@W2[p]) + (b1[p]@W2[p] + b2[p])
// ---------------------------------------------------------------------------
#define NB 65536
#define ND 128
#define NH 256
#define NO 8
#define NE 8
#define EO (NE * NO)        // 64 fused output columns (e*8+o)

typedef __attribute__((ext_vector_type(2))) float v2f;
typedef __attribute__((ext_vector_type(4))) float v4f;
typedef __attribute__((ext_vector_type(8))) float v8f;

// Workspace layout (floats):
//   ws2   [ND/2 * NH * 2]   = 32768  : Ws pre-swizzled into WMMA-B-pair layout
//   weff2 [NH/2 * EO * 2]   = 16384  : folded head weights, same pair layout
//   beff  [EO]              = 64
#define WS2_FLOATS   (ND * NH)
#define WEFF2_FLOATS (NH * EO)

// ---------------------------------------------------------------------------
// B-pair layout for V_WMMA_F32_16X16X4_F32:
// a K-step g needs rows k0 = 4g + 2*lh (+1) at column n (lh = lane>=16).
// We store, per (lh, kpair kp, n):  v4f = { W[k0][n],W[k0+1][n],  (g=2kp)
//                                           W[k0'][n],W[k0'+1][n] } (g=2kp+1)
// so ONE global_load_b128 per lane feeds TWO WMMAs.
// v4f index = (lh*GP + kp)*NCOL + n      (GP = #kpairs, NCOL = columns)
// ---------------------------------------------------------------------------

// Kernel 1a: swizzle Ws (128x256) into ws2. One v2f per thread (16384).
__global__ void qnet_pair_ws(const float* __restrict__ Ws,
                             float* __restrict__ ws2) {
    const int d = blockIdx.x * blockDim.x + threadIdx.x;   // v2f index, 0..16383
    const int j   = d & 1;                 // which K-step inside the pair
    int r         = d >> 1;
    const int n   = r & (NH - 1);          // column 0..255
    r >>= 8;
    const int kp  = r & 15;                // K-pair 0..15
    const int lh  = r >> 4;                // lane half
    const int k0  = 4 * (2 * kp + j) + 2 * lh;
    v2f v;
    v.x = Ws[(size_t)k0 * NH + n];
    v.y = Ws[(size_t)k0 * NH + n + NH];
    ((v2f*)ws2)[d] = v;
}

// Kernel 1b: fold the two routed linears (linear-linear -> one matrix) and
// emit directly in the B-pair layout.   Weff[h][e*8+o] = sum_k W1[e][h][k]*W2[e][k][o]
__global__ void qnet_prep_weff(const float* __restrict__ W1,
                               const float* __restrict__ b1,
                               const float* __restrict__ W2,
                               const float* __restrict__ b2,
                               float* __restrict__ weff2,
                               float* __restrict__ beff) {
    const int tid = blockIdx.x * blockDim.x + threadIdx.x;   // 0..16383
    const int h   = tid >> 6;        // input-feature row (0..255)  == K of GEMM2
    const int col = tid & 63;        // e*8+o
    const int e   = col >> 3;
    const int o   = col & 7;

    const float* w1row = W1 + ((size_t)e * NH + h) * NH;     // W1[e][h][*]
    const float* w2col = W2 + (size_t)e * NH * NO + o;       // W2[e][*][o]
    float acc = 0.0f;
    for (int k = 0; k < NH; ++k)
        acc = __builtin_fmaf(w1row[k], w2col[(size_t)k * NO], acc);

    // scatter into pair layout: h = 4*(2*kp + j) + 2*lh + i
    const int i  = h & 1;
    const int lh = (h >> 1) & 1;
    const int ks = h >> 2;
    const int kp = ks >> 1;
    const int j  = ks & 1;
    weff2[((((size_t)lh * 32 + kp) * EO + col) * 2 + j) * 2 + i] = acc;

    if (tid < EO) {
        const float* b1e = b1 + (size_t)e * NH;
        float bacc = b2[(size_t)e * NO + o];
        for (int k = 0; k < NH; ++k)
            bacc = __builtin_fmaf(b1e[k], w2col[(size_t)k * NO], bacc);
        beff[col] = bacc;
    }
}

// ---------------------------------------------------------------------------
// Kernel 2: fused, both GEMMs in V_WMMA_F32_16X16X4_F32 (fp32-faithful).
// 256 threads = 8 waves, 16 rows per wave, 128 rows/WG, 512 WGs.
// No barriers; LDS holds only wave-private h strips (stride 68 -> 0-conflict).
// ---------------------------------------------------------------------------
#define WG_THREADS 256
#define WG_WAVES   8
#define ROWS_WG    128
#define SH_STR     68

__launch_bounds__(WG_THREADS, 1)
__global__ void qnet_fused(const float* __restrict__ x,
                           const int*   __restrict__ phases,
                           const float* __restrict__ bs,
                           const float* __restrict__ ws2,    // paired Ws
                           const float* __restrict__ weff2,  // paired Weff
                           const float* __restrict__ beff,   // [64]
                           float* __restrict__ out) {        // [B][8]
    __shared__ float sH[WG_WAVES][16 * SH_STR];    // 34816 B

    const int tid   = threadIdx.x;
    const int wave  = tid >> 5;
    const int lane  = tid & 31;
    const int lhalf = lane >> 4;
    const int lmod  = lane & 15;
    const int r0    = blockIdx.x * ROWS_WG + wave * 16;

    const v4f* Bws = (const v4f*)ws2;     // [(lh*16 + kp)*256 + n]
    const v4f* Bwe = (const v4f*)weff2;   // [(lh*32 + kp)*64  + n]

    // ---- A strip: lane's row of x in WMMA 16x4-f32 A layout ---------------
    v2f a[32];
    {
        const float* xrow = x + (size_t)(r0 + lmod) * ND + 2 * lhalf;
        #pragma unroll
        for (int kk = 0; kk < 32; ++kk)
            a[kk] = *(const v2f*)(xrow + 4 * kk);
    }

    float* myH = &sH[wave][0];            // wave-private: DS ops stay in order
    v8f y[4] = {};                        // routed-GEMM accumulators (16 x 64)

    for (int ch = 0; ch < 4; ++ch) {      // 4 column-chunks of h (64 cols each)
        // ================= trunk GEMM chunk, 1-deep B prefetch =============
        v8f c[4] = {};
        v4f bw[4];
        #pragma unroll
        for (int t = 0; t < 4; ++t)
            bw[t] = Bws[(size_t)(lhalf * 16 + 0) * NH + (ch * 4 + t) * 16 + lmod];
        #pragma unroll
        for (int kp = 0; kp < 16; ++kp) {
            v4f bn[4];
            if (kp < 15) {
                #pragma unroll
                for (int t = 0; t < 4; ++t)
                    bn[t] = Bws[(size_t)(lhalf * 16 + kp + 1) * NH +
                                (ch * 4 + t) * 16 + lmod];
            }
            #pragma unroll
            for (int t = 0; t < 4; ++t) {
                const v2f blo = __builtin_shufflevector(bw[t], bw[t], 0, 1);
                c[t] = __builtin_amdgcn_wmma_f32_16x16x4_f32(
                           false, a[2 * kp], false, blo, (short)0, c[t], false, false);
            }
            #pragma unroll
            for (int t = 0; t < 4; ++t) {
                const v2f bhi = __builtin_shufflevector(bw[t], bw[t], 2, 3);
                c[t] = __builtin_amdgcn_wmma_f32_16x16x4_f32(
                           false, a[2 * kp + 1], false, bhi, (short)0, c[t], false, false);
            }
            if (kp < 15) {
                #pragma unroll
                for (int t = 0; t < 4; ++t) bw[t] = bn[t];
            }
        }
        // ---- bias + ReLU on C-layout regs, park chunk in LDS --------------
        #pragma unroll
        for (int t = 0; t < 4; ++t) {
            const float bsv = bs[(ch * 4 + t) * 16 + lmod];
            #pragma unroll
            for (int v = 0; v < 8; ++v) {
                const float hv = fmaxf(c[t][v] + bsv, 0.0f);
                myH[(v + 8 * lhalf) * SH_STR + t * 16 + lmod] = hv;
            }
        }
        // ================= routed GEMM chunk (y += h_chunk @ Weff) =========
        const float* hA = myH + lmod * SH_STR + 2 * lhalf;
        v4f ww[4];
        #pragma unroll
        for (int t = 0; t < 4; ++t)
            ww[t] = Bwe[(size_t)(lhalf * 32 + ch * 8 + 0) * EO + t * 16 + lmod];
        #pragma unroll
        for (int kp = 0; kp < 8; ++kp) {
            v4f wn[4];
            if (kp < 7) {
                #pragma unroll
                for (int t = 0; t < 4; ++t)
                    wn[t] = Bwe[(size_t)(lhalf * 32 + ch * 8 + kp + 1) * EO +
                                t * 16 + lmod];
            }
            const v2f a0 = *(const v2f*)(hA + 4 * (2 * kp));
            const v2f a1 = *(const v2f*)(hA + 4 * (2 * kp + 1));
            #pragma unroll
            for (int t = 0; t < 4; ++t) {
                const v2f blo = __builtin_shufflevector(ww[t], ww[t], 0, 1);
                y[t] = __builtin_amdgcn_wmma_f32_16x16x4_f32(
                           false, a0, false, blo, (short)0, y[t], false, false);
            }
            #pragma unroll
            for (int t = 0; t < 4; ++t) {
                const v2f bhi = __builtin_shufflevector(ww[t], ww[t], 2, 3);
                y[t] = __builtin_amdgcn_wmma_f32_16x16x4_f32(
                           false, a1, false, bhi, (short)0, y[t], false, false);
            }
            if (kp < 7) {
                #pragma unroll
                for (int t = 0; t < 4; ++t) ww[t] = wn[t];
            }
        }
    }

    // ---- spill dense y (16 rows x 64 head-cols) to LDS, routed select -----
    #pragma unroll
    for (int t = 0; t < 4; ++t)
        #pragma unroll
        for (int v = 0; v < 8; ++v)
            myH[(v + 8 * lhalf) * SH_STR + t * 16 + lmod] = y[t][v];

    // 16 rows x 8 outputs per wave -> one v4f quad per lane
    const int selRow = lane >> 1;
    const int oq     = (lane & 1) * 4;
    const int p      = phases[r0 + selRow];
    const v4f yv = *(const v4f*)(myH + selRow * SH_STR + p * NO + oq);
    const v4f bv = *(const v4f*)(beff + p * NO + oq);
    *(v4f*)(out + (size_t)(r0 + selRow) * NO + oq) = yv + bv;
}

// ---------------------------------------------------------------------------
// Entry point
// ---------------------------------------------------------------------------
extern "C" void kernel_launch(void* const* d_in, const int* in_sizes, int n_in,
                              void* d_out, int out_size, void* d_ws, size_t ws_size,
                              hipStream_t stream) {
    const float* x      = (const float*)d_in[0];
    const int*   phases = (const int*)d_in[1];
    const float* Ws     = (const float*)d_in[2];
    const float* bs     = (const float*)d_in[3];
    const float* W1     = (const float*)d_in[4];
    const float* b1     = (const float*)d_in[5];
    const float* W2     = (const float*)d_in[6];
    const float* b2     = (const float*)d_in[7];
    float* out   = (float*)d_out;
    float* ws2   = (float*)d_ws;                 // 32768 floats
    float* weff2 = ws2 + WS2_FLOATS;             // 16384 floats
    float* beff  = weff2 + WEFF2_FLOATS;         // 64 floats

    qnet_pair_ws<<<64, 256, 0, stream>>>(Ws, ws2);
    qnet_prep_weff<<<64, 256, 0, stream>>>(W1, b1, W2, b2, weff2, beff);
    qnet_fused<<<NB / ROWS_WG, WG_THREADS, 0, stream>>>(
        x, phases, bs, ws2, weff2, beff, out);
}